// BTSNet_49581102465696
// MI455X (gfx1250) — compile-verified
//
#include <hip/hip_runtime.h>
#include <math.h>

typedef __attribute__((ext_vector_type(2))) float v2f;
typedef __attribute__((ext_vector_type(8))) float v8f;

#define BB   4
#define PP   65536
#define CCH  64
#define HH   192
#define WW   640
#define NVF  1
#define NVC  4
#define DIN  103      // 64 + 39
#define KPAD 104      // K padded to multiple of 4
#define K2N  (KPAD/2) // 52 pair-rows in W1pair
#define LDSS 106      // padded LDS row stride (even, avoids bank conflicts)
#define EPSV 0.001f
#define DMINV 3.0f
#define DMAXV 80.0f

__device__ __forceinline__ float softplus_f(float x) {
  // jax.nn.softplus = logaddexp(x, 0) = max(x,0) + log1p(exp(-|x|))
  return fmaxf(x, 0.f) + log1pf(expf(-fabsf(x)));
}

struct Bilin { int x0, x1, y0, y1; float w00, w01, w10, w11; };

__device__ __forceinline__ Bilin bilin_setup(float xn, float yn) {
  float xp = (xn + 1.f) * (WW * 0.5f) - 0.5f;
  float yp = (yn + 1.f) * (HH * 0.5f) - 0.5f;
  xp = fminf(fmaxf(xp, 0.f), (float)(WW - 1));
  yp = fminf(fmaxf(yp, 0.f), (float)(HH - 1));
  float xf = floorf(xp), yf = floorf(yp);
  float wx = xp - xf, wy = yp - yf;
  Bilin b;
  b.x0 = (int)xf; b.y0 = (int)yf;
  b.x1 = b.x0 + 1; if (b.x1 > WW - 1) b.x1 = WW - 1;
  b.y1 = b.y0 + 1; if (b.y1 > HH - 1) b.y1 = HH - 1;
  b.w00 = (1.f - wx) * (1.f - wy);
  b.w01 = wx * (1.f - wy);
  b.w10 = (1.f - wx) * wy;
  b.w11 = wx * wy;
  return b;
}

__device__ __forceinline__ void project_pt(const float* __restrict__ K9,
                                           const float* __restrict__ P16,
                                           float x, float y, float z,
                                           float& xn, float& yn, float& zv, bool& inval) {
  float c0 = P16[0] * x + P16[1] * y + P16[2]  * z + P16[3];
  float c1 = P16[4] * x + P16[5] * y + P16[6]  * z + P16[7];
  float c2 = P16[8] * x + P16[9] * y + P16[10] * z + P16[11];
  float p0 = K9[0] * c0 + K9[1] * c1 + K9[2] * c2;
  float p1 = K9[3] * c0 + K9[4] * c1 + K9[5] * c2;
  float p2 = K9[6] * c0 + K9[7] * c1 + K9[8] * c2;
  zv = p2;
  float den = fmaxf(p2, EPSV);
  xn = p0 / den;
  yn = p1 / den;
  inval = (p2 <= EPSV) || (fabsf(xn) > 1.f) || (fabsf(yn) > 1.f);
}

// ---- layout transposes: NCHW -> NHWC (output-index coalesced) ----
__global__ void bts_tfeat(const float* __restrict__ in, float* __restrict__ out) {
  size_t i = (size_t)blockIdx.x * 256 + threadIdx.x;
  const size_t total = (size_t)BB * NVF * HH * WW * CCH;
  if (i >= total) return;
  int c = (int)(i % CCH); size_t t = i / CCH;
  int w = (int)(t % WW);  t /= WW;
  int h = (int)(t % HH);  size_t n = t / HH;
  out[i] = in[(((n * CCH + c) * HH + h) * (size_t)WW) + w];
}

__global__ void bts_tcol(const float* __restrict__ in, float* __restrict__ out) {
  size_t i = (size_t)blockIdx.x * 256 + threadIdx.x;
  const size_t total = (size_t)BB * NVC * HH * WW * 3;
  if (i >= total) return;
  int c = (int)(i % 3);  size_t t = i / 3;
  int w = (int)(t % WW); t /= WW;
  int h = (int)(t % HH); size_t n = t / HH;
  out[i] = in[(((n * 3 + c) * HH + h) * (size_t)WW) + w];
}

// ---- build zero-padded, pair-interleaved W1: out[k2*128+n] = {W1[2k2][n], W1[2k2+1][n]} ----
__global__ void bts_padW1(const float* __restrict__ W1, float* __restrict__ out) {
  int i = blockIdx.x * 256 + threadIdx.x;        // over K2N*128 = 6656
  if (i >= K2N * 128) return;
  int k2 = i >> 7, n = i & 127;
  float w0 = W1[(2 * k2) * 128 + n];                               // 2k2 <= 102, in range
  float w1 = (2 * k2 + 1 < DIN) ? W1[(2 * k2 + 1) * 128 + n] : 0.f;
  out[2 * i + 0] = w0;
  out[2 * i + 1] = w1;
}

// ---- fused: project + gather + posenc (to LDS) + f32 WMMA MLP + color path ----
template <bool CHLAST, bool PADW>
__global__ __launch_bounds__(128) void bts_fused(
    const float* __restrict__ xyz,  const float* __restrict__ feat,
    const float* __restrict__ fK,   const float* __restrict__ fP,
    const float* __restrict__ col,  const float* __restrict__ cK,
    const float* __restrict__ cP,   const float* __restrict__ empty,
    const float* __restrict__ W1,   const float* __restrict__ b1,
    const float* __restrict__ W2,   const float* __restrict__ b2,
    float* __restrict__ rgb_out, float* __restrict__ inv_out, float* __restrict__ sig_out)
{
  __shared__ float smem[128 * LDSS];   // 54,272 B

  const int tid  = threadIdx.x;
  const int lane = tid & 31;
  const int wave = tid >> 5;
  const int b    = blockIdx.x >> 9;            // PP/128 = 512 blocks per batch
  const int p0   = (blockIdx.x & 511) * 128;
  const int p    = p0 + tid;

  const float x = xyz[((size_t)b * PP + p) * 3 + 0];
  const float y = xyz[((size_t)b * PP + p) * 3 + 1];
  const float z = xyz[((size_t)b * PP + p) * 3 + 2];

  // ---- feature view projection (nvf == 1) ----
  float xn, yn, zf; bool invF;
  project_pt(fK + (size_t)b * 9, fP + (size_t)b * 16, x, y, z, xn, yn, zf, invF);
  Bilin bf = bilin_setup(xn, yn);

  float* row = &smem[tid * LDSS];

  if (CHLAST) {
    const size_t base = ((size_t)b * HH) * WW * CCH;
    const float* t00 = feat + base + ((size_t)bf.y0 * WW + bf.x0) * CCH;
    const float* t01 = feat + base + ((size_t)bf.y0 * WW + bf.x1) * CCH;
    const float* t10 = feat + base + ((size_t)bf.y1 * WW + bf.x0) * CCH;
    const float* t11 = feat + base + ((size_t)bf.y1 * WW + bf.x1) * CCH;
    for (int c = 0; c < CCH; c += 4) {
      float4 a = *(const float4*)(t00 + c);
      float4 bq = *(const float4*)(t01 + c);
      float4 cq = *(const float4*)(t10 + c);
      float4 dq = *(const float4*)(t11 + c);
      float4 r4;
      r4.x = bf.w00 * a.x + bf.w01 * bq.x + bf.w10 * cq.x + bf.w11 * dq.x;
      r4.y = bf.w00 * a.y + bf.w01 * bq.y + bf.w10 * cq.y + bf.w11 * dq.y;
      r4.z = bf.w00 * a.z + bf.w01 * bq.z + bf.w10 * cq.z + bf.w11 * dq.z;
      r4.w = bf.w00 * a.w + bf.w01 * bq.w + bf.w10 * cq.w + bf.w11 * dq.w;
      if (invF) r4 = *(const float4*)(empty + c);
      row[c + 0] = r4.x; row[c + 1] = r4.y; row[c + 2] = r4.z; row[c + 3] = r4.w;
    }
  } else {
    for (int c = 0; c < CCH; ++c) {
      const size_t base = (((size_t)b * CCH + c) * HH) * WW;
      float v00 = feat[base + (size_t)bf.y0 * WW + bf.x0];
      float v01 = feat[base + (size_t)bf.y0 * WW + bf.x1];
      float v10 = feat[base + (size_t)bf.y1 * WW + bf.x0];
      float v11 = feat[base + (size_t)bf.y1 * WW + bf.x1];
      float r = bf.w00 * v00 + bf.w01 * v01 + bf.w10 * v10 + bf.w11 * v11;
      if (invF) r = empty[c];
      row[c] = r;
    }
  }

  // ---- positional encoding ----
  const float den = fmaxf(zf, EPSV);
  float zc = (1.f / den - 1.f / DMAXV) / (1.f / DMINV - 1.f / DMAXV);
  zc = 2.f * zc - 1.f;
  row[CCH + 0] = xn; row[CCH + 1] = yn; row[CCH + 2] = zc;
#pragma unroll
  for (int f = 0; f < 6; ++f) {
    const float fr = 1.5f * (float)(1 << f);
    const int o = CCH + 3 + f * 6;
    row[o + 0] = sinf(fr * xn); row[o + 1] = sinf(fr * yn); row[o + 2] = sinf(fr * zc);
    row[o + 3] = cosf(fr * xn); row[o + 4] = cosf(fr * yn); row[o + 5] = cosf(fr * zc);
  }
  row[CCH + 39] = 0.f; row[CCH + 40] = 0.f; row[CCH + 41] = 0.f;   // K padding 103..105

  __syncthreads();

  // ---- MLP layer 1 via V_WMMA_F32_16X16X4_F32 (both M-tiles per B fragment) ----
  const int half = lane >> 4;       // A/B fragment half (K pair select)
  const int lm   = lane & 15;
  float bias1[8], w2v[8];
#pragma unroll
  for (int nt = 0; nt < 8; ++nt) {
    bias1[nt] = b1[nt * 16 + lm];
    w2v[nt]   = W2[nt * 16 + lm];
  }
  const float b2v = b2[0];

  v8f acc[2][8] = {};
  const float* a0base = &smem[(wave * 32 + lm) * LDSS];        // M-tile 0 row
  const float* a1base = a0base + 16 * LDSS;                    // M-tile 1 row

  for (int kt = 0; kt < 26; ++kt) {
    const int kA = kt * 4 + half * 2;            // even -> 8B-aligned LDS b64 load
    v2f A0 = *(const v2f*)(a0base + kA);
    v2f A1 = *(const v2f*)(a1base + kA);
#pragma unroll
    for (int nt = 0; nt < 8; ++nt) {
      v2f Bv;
      if (PADW) {
        // W1 is pre-padded + pair-interleaved: one coalesced b64 load, no guards
        Bv = ((const v2f*)W1)[(size_t)(2 * kt + half) * 128 + nt * 16 + lm];
      } else {
        const int n = nt * 16 + lm;
        Bv.x = W1[(size_t)kA * 128 + n];                       // kA <= 102, in range
        Bv.y = (kA + 1 < DIN) ? W1[(size_t)(kA + 1) * 128 + n] : 0.f;
      }
      acc[0][nt] = __builtin_amdgcn_wmma_f32_16x16x4_f32(
          false, A0, false, Bv, (short)0, acc[0][nt], false, false);
      acc[1][nt] = __builtin_amdgcn_wmma_f32_16x16x4_f32(
          false, A1, false, Bv, (short)0, acc[1][nt], false, false);
    }
  }

  // ---- layer 2: h = relu(acc + b1); sigma = softplus(h . W2 + b2) ----
  // C layout: VGPR r, lanes 0-15 -> M=r, N=lane; lanes 16-31 -> M=r+8, N=lane-16
#pragma unroll
  for (int mt = 0; mt < 2; ++mt) {
#pragma unroll
    for (int r = 0; r < 8; ++r) {
      float s = 0.f;
#pragma unroll
      for (int nt = 0; nt < 8; ++nt)
        s += fmaxf(acc[mt][nt][r] + bias1[nt], 0.f) * w2v[nt];
      s += __shfl_xor(s, 1, 32);
      s += __shfl_xor(s, 2, 32);
      s += __shfl_xor(s, 4, 32);
      s += __shfl_xor(s, 8, 32);     // halves stay separate (offsets < 16)
      if (lm == r) {
        const int M = r + half * 8;
        const int pOut = p0 + wave * 32 + mt * 16 + M;
        sig_out[(size_t)b * PP + pOut] = softplus_f(s + b2v);
      }
    }
  }

  // ---- color views ----
#pragma unroll
  for (int v = 0; v < NVC; ++v) {
    float xc, yc, zcv; bool invC;
    project_pt(cK + ((size_t)b * NVC + v) * 9, cP + ((size_t)b * NVC + v) * 16,
               x, y, z, xc, yc, zcv, invC);
    Bilin bc = bilin_setup(xc, yc);
    float r3[3];
    if (CHLAST) {
      const size_t base = ((size_t)(b * NVC + v) * HH) * WW * 3;
      const float* t00 = col + base + ((size_t)bc.y0 * WW + bc.x0) * 3;
      const float* t01 = col + base + ((size_t)bc.y0 * WW + bc.x1) * 3;
      const float* t10 = col + base + ((size_t)bc.y1 * WW + bc.x0) * 3;
      const float* t11 = col + base + ((size_t)bc.y1 * WW + bc.x1) * 3;
#pragma unroll
      for (int c = 0; c < 3; ++c)
        r3[c] = bc.w00 * t00[c] + bc.w01 * t01[c] + bc.w10 * t10[c] + bc.w11 * t11[c];
    } else {
#pragma unroll
      for (int c = 0; c < 3; ++c) {
        const size_t base = (((size_t)(b * NVC + v) * 3 + c) * HH) * WW;
        float v00 = col[base + (size_t)bc.y0 * WW + bc.x0];
        float v01 = col[base + (size_t)bc.y0 * WW + bc.x1];
        float v10 = col[base + (size_t)bc.y1 * WW + bc.x0];
        float v11 = col[base + (size_t)bc.y1 * WW + bc.x1];
        r3[c] = bc.w00 * v00 + bc.w01 * v01 + bc.w10 * v10 + bc.w11 * v11;
      }
    }
    const size_t ob = ((size_t)b * PP + p) * 12 + v * 3;
    rgb_out[ob + 0] = r3[0];
    rgb_out[ob + 1] = r3[1];
    rgb_out[ob + 2] = r3[2];
    inv_out[((size_t)b * PP + p) * 4 + v] = (invC || invF) ? 1.f : 0.f;
  }
}

extern "C" void kernel_launch(void* const* d_in, const int* in_sizes, int n_in,
                              void* d_out, int out_size, void* d_ws, size_t ws_size,
                              hipStream_t stream) {
  const float* xyz   = (const float*)d_in[0];
  const float* gf    = (const float*)d_in[1];
  const float* fK    = (const float*)d_in[2];
  const float* fP    = (const float*)d_in[3];
  const float* gc    = (const float*)d_in[4];
  const float* cK    = (const float*)d_in[5];
  const float* cP    = (const float*)d_in[6];
  const float* empty = (const float*)d_in[7];
  const float* W1    = (const float*)d_in[8];
  const float* b1    = (const float*)d_in[9];
  const float* W2    = (const float*)d_in[10];
  const float* b2    = (const float*)d_in[11];

  float* rgb = (float*)d_out;
  float* inv = rgb + (size_t)BB * PP * 12;
  float* sig = inv + (size_t)BB * PP * 4;

  const size_t featBytes = (size_t)BB * NVF * HH * WW * CCH * sizeof(float);
  const size_t colBytes  = (size_t)BB * NVC * HH * WW * 3 * sizeof(float);
  const size_t w1pBytes  = (size_t)K2N * 128 * 2 * sizeof(float);

  const int nBlocks   = BB * (PP / 128);
  const int padBlocks = (K2N * 128 + 255) / 256;

  if (ws_size >= featBytes + colBytes + w1pBytes) {
    // tier A: transposed grids + padded weights
    float* featT = (float*)d_ws;
    float* colT  = (float*)((char*)d_ws + featBytes);
    float* w1p   = (float*)((char*)d_ws + featBytes + colBytes);
    {
      const size_t total = (size_t)BB * NVF * HH * WW * CCH;
      bts_tfeat<<<(int)((total + 255) / 256), 256, 0, stream>>>(gf, featT);
    }
    {
      const size_t total = (size_t)BB * NVC * HH * WW * 3;
      bts_tcol<<<(int)((total + 255) / 256), 256, 0, stream>>>(gc, colT);
    }
    bts_padW1<<<padBlocks, 256, 0, stream>>>(W1, w1p);
    bts_fused<true, true><<<nBlocks, 128, 0, stream>>>(
        xyz, featT, fK, fP, colT, cK, cP, empty, w1p, b1, W2, b2, rgb, inv, sig);
  } else if (ws_size >= w1pBytes) {
    // tier B: padded weights only, original (CHW) grid layouts
    float* w1p = (float*)d_ws;
    bts_padW1<<<padBlocks, 256, 0, stream>>>(W1, w1p);
    bts_fused<false, true><<<nBlocks, 128, 0, stream>>>(
        xyz, gf, fK, fP, gc, cK, cP, empty, w1p, b1, W2, b2, rgb, inv, sig);
  } else {
    // tier C: no workspace at all — guarded weight loads
    bts_fused<false, false><<<nBlocks, 128, 0, stream>>>(
        xyz, gf, fK, fP, gc, cK, cP, empty, W1, b1, W2, b2, rgb, inv, sig);
  }
}